// BTCVAELoss_78795470012510
// MI455X (gfx1250) — compile-verified
//
#include <hip/hip_runtime.h>
#include <math.h>

// ---------------------------------------------------------------------------
// BTC-VAE loss, fp32, B=1024, D=64.
//   recon      : memory-bound float4 reduction (two-stage, deterministic)
//   pair matrix: single fp32 WMMA GEMM [1024x128]x[128x1024]
//                (V_WMMA_F32_16X16X4_F32, k-pair-interleaved operands so every
//                 fragment load is one coalesced global_load_b64 per lane)
//   log_prod   : 67M-element online logsumexp, float4-packed LDS tiles
// ---------------------------------------------------------------------------

typedef __attribute__((ext_vector_type(2))) float v2f;
typedef __attribute__((ext_vector_type(8))) float v8f;

#define B_SZ   1024
#define D_SZ   64
#define KTOT   128            // concat: [-0.5*z^2 | z]  x  [w | mu*w]
#define TJ     32             // j-tile for the LSE kernel

// workspace layout (in floats)
#define OFF_RPART  0                          // 4096 recon per-block partials
#define OFF_Z      (OFF_RPART + 4096)         // [B,D]
#define OFF_CPOST  (OFF_Z     + B_SZ*D_SZ)    // [B]   c_j - 0.5*v_j
#define OFF_LQC    (OFF_CPOST + B_SZ)         // [B]   log q(z|x)
#define OFF_LPZ    (OFF_LQC   + B_SZ)         // [B]   log p(z)
#define OFF_X2     (OFF_LPZ   + B_SZ)         // [K/2][B*2] interleaved A op
#define OFF_Y2     (OFF_X2    + B_SZ*KTOT)    // [K/2][B*2] interleaved B op
#define OFF_PACK   (OFF_Y2    + B_SZ*KTOT)    // [B*D] float4 (alog, mu, w, 0)
#define OFF_M2     (OFF_PACK  + B_SZ*D_SZ*4)  // [B,B] pair matrix + log_iw
#define OFF_LQZ    (OFF_M2    + B_SZ*B_SZ)    // [B]
#define OFF_LPQ    (OFF_LQZ   + B_SZ)         // [B]

#define LOG2PI_F 1.8378770664093453f

__device__ __forceinline__ float liw_sel(int i, int j,
                                         float l_def, float l_diag, float l_strat) {
    // W[:, 1] = strat; W[B-2, 0] = strat; diag = 1/N; else 1/(B-1)
    float r = (i == j) ? l_diag : l_def;
    r = (i == (B_SZ - 2) && j == 0) ? l_strat : r;
    r = (j == 1) ? l_strat : r;
    return r;
}

// ---------------- K1: reconstruction MSE partials (memory bound) -----------
__global__ __launch_bounds__(256) void k_recon(const float4* __restrict__ rx,
                                               const float4* __restrict__ x,
                                               float* __restrict__ partial, int n4) {
    __shared__ float sr[256];
    float acc = 0.f;
    for (int idx = blockIdx.x * 256 + threadIdx.x; idx < n4; idx += gridDim.x * 256) {
        float4 a = rx[idx], b = x[idx];
        float d0 = a.x - b.x, d1 = a.y - b.y, d2 = a.z - b.z, d3 = a.w - b.w;
        acc += d0 * d0 + d1 * d1 + d2 * d2 + d3 * d3;
    }
    sr[threadIdx.x] = acc;
    __syncthreads();
    for (int off = 128; off >= 1; off >>= 1) {
        if (threadIdx.x < off) sr[threadIdx.x] += sr[threadIdx.x + off];
        __syncthreads();
    }
    if (threadIdx.x == 0) partial[blockIdx.x] = sr[0];
}

// ---------------- K2: per-row prep (z, operands, packed tile, row stats) ---
__global__ __launch_bounds__(64) void k_prep(const float* __restrict__ mu,
                                             const float* __restrict__ logvar,
                                             const float* __restrict__ noise,
                                             float* __restrict__ ws) {
    __shared__ float sr[4 * 64];
    const int i = blockIdx.x;
    const int d = threadIdx.x;
    const int id = i * D_SZ + d;

    const float lv = logvar[id];
    const float m  = mu[id];
    const float nz = noise[id];

    const float w    = __expf(-lv);
    const float z    = m + nz * __expf(0.5f * lv);
    const float alog = -0.5f * (LOG2PI_F + lv);

    ws[OFF_Z + id] = z;

    // k-pair interleaved transposed operands: elem (row, k) at [k/2][row*2+(k&1)]
    const int kk  = d >> 1;          // k-pair index for k = d   (0..31)
    const int par = d & 1;
    const int col2 = i * 2 + par;
    ws[OFF_X2 + (kk)      * (2 * B_SZ) + col2] = -0.5f * z * z;  // k in [0,64)
    ws[OFF_X2 + (kk + 32) * (2 * B_SZ) + col2] = z;              // k in [64,128)
    ws[OFF_Y2 + (kk)      * (2 * B_SZ) + col2] = w;
    ws[OFF_Y2 + (kk + 32) * (2 * B_SZ) + col2] = m * w;

    ((float4*)(ws + OFF_PACK))[id] = make_float4(alog, m, w, 0.f);

    sr[d]       = alog;                               // -> c_j
    sr[64 + d]  = m * m * w;                          // -> v_j
    sr[128 + d] = alog - 0.5f * nz * nz;              // -> log q(z|x)
    sr[192 + d] = -0.5f * LOG2PI_F - 0.5f * z * z;    // -> log p(z)
    __syncthreads();
    for (int off = 32; off >= 1; off >>= 1) {
        if (d < off) {
            sr[d]       += sr[d + off];
            sr[64 + d]  += sr[64 + d + off];
            sr[128 + d] += sr[128 + d + off];
            sr[192 + d] += sr[192 + d + off];
        }
        __syncthreads();
    }
    if (d == 0) {
        ws[OFF_CPOST + i] = sr[0] - 0.5f * sr[64];
        ws[OFF_LQC + i]   = sr[128];
        ws[OFF_LPZ + i]   = sr[192];
    }
}

// ---------------- K3: fp32 WMMA GEMM -> M2[i,j] (+ log_iw) -----------------
// One wave per 16x16 tile; K=128 as 32 steps of V_WMMA_F32_16X16X4_F32.
// Fragment loads: one global_load_b64 per lane per operand per step, each
// half-wave covering one contiguous 128B segment of the interleaved layout.
__global__ __launch_bounds__(256) void k_pair_gemm(const float* __restrict__ X2,
                                                   const float* __restrict__ Y2,
                                                   const float* __restrict__ cpost,
                                                   float* __restrict__ M2) {
    const int wave = blockIdx.x * (blockDim.x >> 5) + (threadIdx.x >> 5);
    const int lane = threadIdx.x & 31;
    const int it = wave >> 6;              // 64 row tiles
    const int jt = wave & 63;              // 64 col tiles
    const int i0 = it << 4, j0 = jt << 4;

    const int r  = lane & 15;              // A row / B col within tile
    const int kh = lane >> 4;              // lanes 16-31 hold K+2,K+3 (next k-pair)

    const float* Arow = X2 + (i0 + r) * 2; // + kpair*2048
    const float* Brow = Y2 + (j0 + r) * 2;

    v8f c = {};
#pragma unroll
    for (int k0 = 0; k0 < KTOT; k0 += 4) {
        const int kp = (k0 >> 1) + kh;     // k-pair index for this lane half
        const v2f a = *(const v2f*)(Arow + kp * (2 * B_SZ));
        const v2f b = *(const v2f*)(Brow + kp * (2 * B_SZ));
        c = __builtin_amdgcn_wmma_f32_16x16x4_f32(
                false, a, false, b, (short)0, c, false, false);
    }

    const float l_def   = __logf(1.0f / 1023.0f);
    const float l_diag  = __logf(1.0f / 50000.0f);
    const float l_strat = __logf(48977.0f / 51150000.0f);  // (N-M)/(N*M)

    const int j  = j0 + r;                 // N = lane&15
    const int mo = kh * 8;                 // lanes 16-31 hold M = 8..15
    const float cp = cpost[j];
#pragma unroll
    for (int rr = 0; rr < 8; ++rr) {
        const int i = i0 + mo + rr;
        M2[i * B_SZ + j] = c[rr] + cp + liw_sel(i, j, l_def, l_diag, l_strat);
    }
}

// ---------------- K4: row-wise logsumexp of M2 -> log_qz -------------------
__global__ __launch_bounds__(256) void k_logqz(const float* __restrict__ M2,
                                               float* __restrict__ log_qz) {
    __shared__ float sr[256];
    const int i = blockIdx.x, tid = threadIdx.x;
    const float* row = M2 + i * B_SZ;
    float m = -INFINITY;
    for (int j = tid; j < B_SZ; j += 256) m = fmaxf(m, row[j]);
    sr[tid] = m; __syncthreads();
    for (int off = 128; off >= 1; off >>= 1) {
        if (tid < off) sr[tid] = fmaxf(sr[tid], sr[tid + off]);
        __syncthreads();
    }
    const float gm = sr[0]; __syncthreads();
    float s = 0.f;
    for (int j = tid; j < B_SZ; j += 256) s += __expf(row[j] - gm);
    sr[tid] = s; __syncthreads();
    for (int off = 128; off >= 1; off >>= 1) {
        if (tid < off) sr[tid] += sr[tid + off];
        __syncthreads();
    }
    if (tid == 0) log_qz[i] = gm + __logf(sr[0]);
}

// ---------------- K5: per-(i,d) logsumexp over j, summed over d ------------
// Block = 256 threads = 4 rows of i x 64 dims; (alog,mu,w) float4-packed in LDS:
// one ds_load_b128 per inner iteration, 16B/lane coalesced global staging.
__global__ __launch_bounds__(256) void k_logprod(const float* __restrict__ z,
                                                 const float4* __restrict__ pack,
                                                 float* __restrict__ log_prod) {
    __shared__ float4 s_pack[TJ * D_SZ];   // 32 KB
    __shared__ float  s_red[256];

    const int tid = threadIdx.x;
    const int d   = tid & 63;
    const int i   = blockIdx.x * 4 + (tid >> 6);
    const float zi = z[i * D_SZ + d];

    const float l_def   = __logf(1.0f / 1023.0f);
    const float l_diag  = __logf(1.0f / 50000.0f);
    const float l_strat = __logf(48977.0f / 51150000.0f);

    float m = -INFINITY, s = 0.f;
    for (int t = 0; t < B_SZ; t += TJ) {
        __syncthreads();
        for (int u = tid; u < TJ * D_SZ; u += 256)
            s_pack[u] = pack[t * D_SZ + u];
        if (t + TJ < B_SZ)   // hint next tile into cache (global_prefetch_b8)
            __builtin_prefetch(&pack[(t + TJ) * D_SZ + tid * 8], 0, 1);
        __syncthreads();
#pragma unroll 4
        for (int jj = 0; jj < TJ; ++jj) {
            const int j = t + jj;
            const float4 p = s_pack[jj * D_SZ + d];   // (alog, mu, w, -)
            const float dz = zi - p.y;
            float e = p.x - 0.5f * dz * dz * p.z
                      + liw_sel(i, j, l_def, l_diag, l_strat);
            const float nm = fmaxf(m, e);
            s = s * __expf(m - nm) + __expf(e - nm);
            m = nm;
        }
    }
    s_red[tid] = m + __logf(s);   // lse for (i,d)
    __syncthreads();
    for (int off = 32; off >= 1; off >>= 1) {
        if (d < off) s_red[tid] += s_red[tid + off];
        __syncthreads();
    }
    if (d == 0) log_prod[i] = s_red[tid];
}

// ---------------- K6: final scalar ----------------------------------------
__global__ __launch_bounds__(256) void k_final(const float* __restrict__ rpart,
                                               const float* __restrict__ lqc,
                                               const float* __restrict__ lpz,
                                               const float* __restrict__ lqz,
                                               const float* __restrict__ lpq,
                                               float* __restrict__ out) {
    __shared__ float s0[256];
    __shared__ float s1[256];
    const int tid = threadIdx.x;
    float a0 = 0.f, a1 = 0.f;
    for (int k = tid; k < 4096; k += 256) a0 += rpart[k];
    // ALPHA*(lqc-lqz) + BETA*(lqz-lpq) + GAMMA*(lpq-lpz), ALPHA=GAMMA=1, BETA=6
    for (int k = tid; k < B_SZ; k += 256)
        a1 += lqc[k] + 5.0f * lqz[k] - 5.0f * lpq[k] - lpz[k];
    s0[tid] = a0; s1[tid] = a1;
    __syncthreads();
    for (int off = 128; off >= 1; off >>= 1) {
        if (tid < off) { s0[tid] += s0[tid + off]; s1[tid] += s1[tid + off]; }
        __syncthreads();
    }
    if (tid == 0) out[0] = s0[0] + s1[0] * (1.0f / (float)B_SZ);
}

// ---------------------------------------------------------------------------
extern "C" void kernel_launch(void* const* d_in, const int* in_sizes, int n_in,
                              void* d_out, int out_size, void* d_ws, size_t ws_size,
                              hipStream_t stream) {
    const float* recon_x = (const float*)d_in[0];
    const float* x       = (const float*)d_in[1];
    const float* mu      = (const float*)d_in[2];
    const float* logvar  = (const float*)d_in[3];
    const float* noise   = (const float*)d_in[4];
    float* ws  = (float*)d_ws;
    float* out = (float*)d_out;

    const int n4 = in_sizes[0] / 4;  // 12582912 / 4

    k_recon<<<4096, 256, 0, stream>>>((const float4*)recon_x, (const float4*)x,
                                      ws + OFF_RPART, n4);

    k_prep<<<B_SZ, 64, 0, stream>>>(mu, logvar, noise, ws);

    // 4096 tiles (64x64), 8 waves/block -> 512 blocks
    k_pair_gemm<<<512, 256, 0, stream>>>(ws + OFF_X2, ws + OFF_Y2,
                                         ws + OFF_CPOST, ws + OFF_M2);

    k_logqz<<<B_SZ, 256, 0, stream>>>(ws + OFF_M2, ws + OFF_LQZ);

    k_logprod<<<B_SZ / 4, 256, 0, stream>>>(ws + OFF_Z,
                                            (const float4*)(ws + OFF_PACK),
                                            ws + OFF_LPQ);

    k_final<<<1, 256, 0, stream>>>(ws + OFF_RPART, ws + OFF_LQC, ws + OFF_LPZ,
                                   ws + OFF_LQZ, ws + OFF_LPQ, out);
}